// SpectralConv3D_41446434406459
// MI455X (gfx1250) — compile-verified
//
#include <hip/hip_runtime.h>
#include <cstddef>
#include <cstdint>

// CDNA5 / gfx1250: wave32, WMMA f32 16x16x4 (native fp32 matrix op).
typedef __attribute__((ext_vector_type(2))) float v2f;
typedef __attribute__((ext_vector_type(8))) float v8f;
typedef __attribute__((ext_vector_type(4))) unsigned int u32x4;
typedef __attribute__((ext_vector_type(8))) int i32x8;
typedef __attribute__((ext_vector_type(4))) int i32x4;

__device__ __forceinline__ v8f wmma4(v2f a, v2f b, v8f c) {
  // D = A(16x4) * B(4x16) + C(16x16), all fp32.
  return __builtin_amdgcn_wmma_f32_16x16x4_f32(false, a, false, b, (short)0, c,
                                               false, false);
}

// Geometry: B=4, spatial N=64 per axis, C=CIN=COUT=32, M=16 modes per axis.
#define TWO_PI_OVER_64 0.09817477042f
#define IFFT_SCALE 3.814697265625e-6f  // 1/64^3

// ---------------------------------------------------------------------------
// Weight transpose: w[i][o][mode] -> wt[mode][i][o]  (mode = kx*256+ky*16+kz)
// ---------------------------------------------------------------------------
__global__ void k_wt_transpose(const float* __restrict__ wr,
                               const float* __restrict__ wi,
                               float* __restrict__ tre,
                               float* __restrict__ tim) {
  const int tid = blockIdx.x * blockDim.x + threadIdx.x;  // 0 .. 4194303
  const int mode = tid & 4095;
  const int c = tid >> 12;  // i*32+o
  const size_t dst = (size_t)mode * 1024 + (size_t)c;
  tre[dst] = wr[tid];
  tim[dst] = wi[tid];
}

// ---------------------------------------------------------------------------
// Stage 1: forward z DFT (real input), TDM-fed.
// Workgroup = 4 waves; one TENSOR_LOAD_TO_LDS brings 4 sites (4 x 8KB) into
// LDS, then each wave computes one site:
//   out[site][kz][i] = sum_z F[kz][z] * x[site][z][i]   (16x64 . 64x32)
// ---------------------------------------------------------------------------
__global__ void k_fwd_z_real(const float* __restrict__ x,
                             float* __restrict__ outre,
                             float* __restrict__ outim) {
  __shared__ __align__(16) float FtabRe[1024];  // [m(16)][z(64)]
  __shared__ __align__(16) float FtabIm[1024];
  __shared__ __align__(16) float tileB[4 * 2048];  // 4 sites x 64z x 32ch

  for (int idx = threadIdx.x; idx < 1024; idx += blockDim.x) {
    const int mm = idx >> 6, zz = idx & 63;
    const int p = (mm * zz) & 63;
    float s, c;
    __sincosf(-TWO_PI_OVER_64 * (float)p, &s, &c);
    FtabRe[idx] = c;
    FtabIm[idx] = s;
  }

  const int site0 = blockIdx.x * 4;
  if ((threadIdx.x >> 5) == 0) {
    // Tensor DMA descriptor (ISA 8.3/8.4): 2-D tensor, data_size=4B,
    // tensor_dim0=2048 elems (one site row = 64z*32ch), 4 rows, stride 2048.
    const unsigned long long ga =
        (unsigned long long)(uintptr_t)(x + (size_t)site0 * 2048);
    const unsigned ldsa = (unsigned)(uintptr_t)(void*)tileB;
    const u32x4 g0 = {
        1u,                                                // count=1 (valid)
        ldsa,                                              // lds_addr
        (unsigned)ga,                                      // global_addr lo
        ((unsigned)(ga >> 32) & 0x1FFFFFFu) | (2u << 30)}; // ga hi | type=2
    const i32x8 g1 = {
        0x20000,      // [17:16] data_size = 2 (4 bytes); no multicast/pad
        2048 << 16,   // tensor_dim0[15:0] in [31:16]
        4 << 16,      // tensor_dim0 hi=0 | tensor_dim1 = 4
        2048 << 16,   // tensor_dim1 hi=0 | tile_dim0 = 2048
        4,            // tile_dim1 = 4, tile_dim2 = 0
        2048,         // tensor_dim0_stride = 2048
        0, 0};
    const i32x4 gz4 = {0, 0, 0, 0};
    const i32x8 gz8 = {0, 0, 0, 0, 0, 0, 0, 0};
    __builtin_amdgcn_tensor_load_to_lds(g0, g1, gz4, gz4, gz8, 0);
    __builtin_amdgcn_s_wait_tensorcnt(0);
  }
  __syncthreads();

  const int wv = threadIdx.x >> 5;
  const int lane = threadIdx.x & 31;
  const int m = lane & 15, half = lane >> 4;
  const float* bx = tileB + wv * 2048;

  v8f ar0 = {}, ar1 = {}, ai0 = {}, ai1 = {};
#pragma unroll
  for (int ks = 0; ks < 16; ++ks) {
    const int z0 = ks * 4 + half * 2;
    const v2f fre = *reinterpret_cast<const v2f*>(&FtabRe[m * 64 + z0]);
    const v2f fim = *reinterpret_cast<const v2f*>(&FtabIm[m * 64 + z0]);
    const float* p0 = bx + z0 * 32;
    v2f b0, b1;
    b0.x = p0[m];      b0.y = p0[32 + m];
    b1.x = p0[16 + m]; b1.y = p0[48 + m];
    ar0 = wmma4(fre, b0, ar0);
    ai0 = wmma4(fim, b0, ai0);
    ar1 = wmma4(fre, b1, ar1);
    ai1 = wmma4(fim, b1, ai1);
  }
  const int site = site0 + wv;
  float* ore = outre + (size_t)site * 512;
  float* oim = outim + (size_t)site * 512;
#pragma unroll
  for (int r = 0; r < 8; ++r) {
    const int kz = r + half * 8;
    ore[kz * 32 + m] = ar0[r];
    ore[kz * 32 + 16 + m] = ar1[r];
    oim[kz * 32 + m] = ai0[r];
    oim[kz * 32 + 16 + m] = ai1[r];
  }
}

// ---------------------------------------------------------------------------
// Generic forward DFT over a length-64 axis, complex input.
// in  : [batch][64][ncols] (re,im planes)   out : [batch][16][ncols]
// Per wave: one (batch, 16-column tile).
// ---------------------------------------------------------------------------
__global__ void k_fwd_dft_cplx(const float* __restrict__ inre,
                               const float* __restrict__ inim,
                               float* __restrict__ outre,
                               float* __restrict__ outim, int ncols,
                               int ntiles) {
  __shared__ __align__(16) float FtabRe[1024];
  __shared__ __align__(16) float FtabIm[1024];
  for (int idx = threadIdx.x; idx < 1024; idx += blockDim.x) {
    const int mm = idx >> 6, zz = idx & 63;
    const int p = (mm * zz) & 63;
    float s, c;
    __sincosf(-TWO_PI_OVER_64 * (float)p, &s, &c);
    FtabRe[idx] = c;
    FtabIm[idx] = s;
  }
  __syncthreads();

  const int wave = blockIdx.x * (blockDim.x >> 5) + (threadIdx.x >> 5);
  const int lane = threadIdx.x & 31;
  const int m = lane & 15, half = lane >> 4;
  const int batch = wave / ntiles;
  const int nt = wave - batch * ntiles;
  const size_t ib = (size_t)batch * 64 * (size_t)ncols + (size_t)nt * 16;
  const size_t ob = (size_t)batch * 16 * (size_t)ncols + (size_t)nt * 16;

  v8f ar = {}, ai = {};
#pragma unroll
  for (int ks = 0; ks < 16; ++ks) {
    const int z0 = ks * 4 + half * 2;
    const v2f fre = *reinterpret_cast<const v2f*>(&FtabRe[m * 64 + z0]);
    const v2f fim = *reinterpret_cast<const v2f*>(&FtabIm[m * 64 + z0]);
    const v2f nfim = -fim;  // F32 WMMA has no A-negate modifier: VALU negate
    const float* pr = inre + ib + (size_t)z0 * ncols;
    const float* pi = inim + ib + (size_t)z0 * ncols;
    v2f br, bi;
    br.x = pr[m]; br.y = pr[ncols + m];
    bi.x = pi[m]; bi.y = pi[ncols + m];
    ar = wmma4(fre, br, ar);
    ar = wmma4(nfim, bi, ar);
    ai = wmma4(fre, bi, ai);
    ai = wmma4(fim, br, ai);
  }
#pragma unroll
  for (int r = 0; r < 8; ++r) {
    const int kk = r + half * 8;
    outre[ob + (size_t)kk * ncols + m] = ar[r];
    outim[ob + (size_t)kk * ncols + m] = ai[r];
  }
}

// ---------------------------------------------------------------------------
// Stage 4: per-mode channel mix.  Y[b][o] = sum_i X[b][i] * W[i][o] (complex)
// X: [b][mode][i] (b-stride 4096*32), Wt: [mode][i][o].  M rows 4..15 zeroed.
// ---------------------------------------------------------------------------
__global__ void k_mode_mix(const float* __restrict__ xre,
                           const float* __restrict__ xim,
                           const float* __restrict__ wre,
                           const float* __restrict__ wim,
                           float* __restrict__ yre, float* __restrict__ yim) {
  const int wave = blockIdx.x * (blockDim.x >> 5) + (threadIdx.x >> 5);
  const int lane = threadIdx.x & 31;
  const int m = lane & 15, half = lane >> 4;
  const int mode = wave;
  const size_t BST = 131072;  // 4096 modes * 32 ch
  const size_t xb = (size_t)mode * 32;
  const float* wr = wre + (size_t)mode * 1024;
  const float* wi = wim + (size_t)mode * 1024;

  v8f cr0 = {}, cr1 = {}, ci0 = {}, ci1 = {};
#pragma unroll
  for (int ks = 0; ks < 8; ++ks) {
    const int k0 = ks * 4 + half * 2;  // i index (even -> 8B aligned pair)
    v2f are = {0.f, 0.f}, aim = {0.f, 0.f};
    if (m < 4) {  // only 4 batch rows exist; rest of M is zero
      are = *reinterpret_cast<const v2f*>(&xre[(size_t)m * BST + xb + k0]);
      aim = *reinterpret_cast<const v2f*>(&xim[(size_t)m * BST + xb + k0]);
    }
    const v2f naim = -aim;
    const float* w0 = wr + k0 * 32;
    const float* w1 = wi + k0 * 32;
    v2f br0, br1, bi0, bi1;
    br0.x = w0[m];      br0.y = w0[32 + m];
    br1.x = w0[16 + m]; br1.y = w0[48 + m];
    bi0.x = w1[m];      bi0.y = w1[32 + m];
    bi1.x = w1[16 + m]; bi1.y = w1[48 + m];
    cr0 = wmma4(are, br0, cr0);  cr0 = wmma4(naim, bi0, cr0);
    ci0 = wmma4(are, bi0, ci0);  ci0 = wmma4(aim, br0, ci0);
    cr1 = wmma4(are, br1, cr1);  cr1 = wmma4(naim, bi1, cr1);
    ci1 = wmma4(are, bi1, ci1);  ci1 = wmma4(aim, br1, ci1);
  }
  if (half == 0) {
#pragma unroll
    for (int r = 0; r < 4; ++r) {  // rows 0..3 == batch index
      const size_t d = (size_t)r * BST + xb;
      yre[d + m] = cr0[r];
      yre[d + 16 + m] = cr1[r];
      yim[d + m] = ci0[r];
      yim[d + 16 + m] = ci1[r];
    }
  }
}

// ---------------------------------------------------------------------------
// Generic inverse DFT expansion 16 modes -> 64 points, complex.
// in : [batch][16][ncols]   out : [batch][64][ncols]
// Per wave: one (batch, 16-row mtile, 16-col ntile).
// ---------------------------------------------------------------------------
__global__ void k_inv_dft_cplx(const float* __restrict__ inre,
                               const float* __restrict__ inim,
                               float* __restrict__ outre,
                               float* __restrict__ outim, int ncols,
                               int ntiles) {
  __shared__ __align__(16) float GtabRe[1024];  // [x(64)][k(16)]
  __shared__ __align__(16) float GtabIm[1024];
  for (int idx = threadIdx.x; idx < 1024; idx += blockDim.x) {
    const int xx = idx >> 4, kk = idx & 15;
    const int p = (xx * kk) & 63;
    float s, c;
    __sincosf(TWO_PI_OVER_64 * (float)p, &s, &c);
    GtabRe[idx] = c;
    GtabIm[idx] = s;
  }
  __syncthreads();

  const int wave = blockIdx.x * (blockDim.x >> 5) + (threadIdx.x >> 5);
  const int lane = threadIdx.x & 31;
  const int m = lane & 15, half = lane >> 4;
  const int perbatch = 4 * ntiles;
  const int batch = wave / perbatch;
  const int rem = wave - batch * perbatch;
  const int mtile = rem / ntiles;
  const int nt = rem - mtile * ntiles;
  const size_t ib = (size_t)batch * 16 * (size_t)ncols + (size_t)nt * 16;
  const size_t ob = (size_t)batch * 64 * (size_t)ncols + (size_t)nt * 16;
  const int xr = mtile * 16 + m;

  v8f ar = {}, ai = {};
#pragma unroll
  for (int ks = 0; ks < 4; ++ks) {
    const int k0 = ks * 4 + half * 2;
    const v2f gre = *reinterpret_cast<const v2f*>(&GtabRe[xr * 16 + k0]);
    const v2f gim = *reinterpret_cast<const v2f*>(&GtabIm[xr * 16 + k0]);
    const v2f ngim = -gim;
    const float* pr = inre + ib + (size_t)k0 * ncols;
    const float* pi = inim + ib + (size_t)k0 * ncols;
    v2f br, bi;
    br.x = pr[m]; br.y = pr[ncols + m];
    bi.x = pi[m]; bi.y = pi[ncols + m];
    ar = wmma4(gre, br, ar);
    ar = wmma4(ngim, bi, ar);
    ai = wmma4(gre, bi, ai);
    ai = wmma4(gim, br, ai);
  }
#pragma unroll
  for (int r = 0; r < 8; ++r) {
    const int xx = mtile * 16 + r + half * 8;
    outre[ob + (size_t)xx * ncols + m] = ar[r];
    outim[ob + (size_t)xx * ncols + m] = ai[r];
  }
}

// ---------------------------------------------------------------------------
// Stage 7: inverse z expansion, take real part, apply 1/64^3 iFFT scale.
// in : [site][kz(16)][o(32)]   out : [site][z(64)][o(32)]
// Per wave: one (site, z-mtile).
// ---------------------------------------------------------------------------
__global__ void k_inv_z_real(const float* __restrict__ inre,
                             const float* __restrict__ inim,
                             float* __restrict__ out) {
  __shared__ __align__(16) float GtabRe[1024];
  __shared__ __align__(16) float GtabIm[1024];
  for (int idx = threadIdx.x; idx < 1024; idx += blockDim.x) {
    const int xx = idx >> 4, kk = idx & 15;
    const int p = (xx * kk) & 63;
    float s, c;
    __sincosf(TWO_PI_OVER_64 * (float)p, &s, &c);
    GtabRe[idx] = c;
    GtabIm[idx] = s;
  }
  __syncthreads();

  const int wave = blockIdx.x * (blockDim.x >> 5) + (threadIdx.x >> 5);
  const int lane = threadIdx.x & 31;
  const int m = lane & 15, half = lane >> 4;
  const int site = wave >> 2;
  const int mtile = wave & 3;
  const float* pr0 = inre + (size_t)site * 512;
  const float* pi0 = inim + (size_t)site * 512;
  const int zr = mtile * 16 + m;

  v8f a0 = {}, a1 = {};
#pragma unroll
  for (int ks = 0; ks < 4; ++ks) {
    const int k0 = ks * 4 + half * 2;
    const v2f gre = *reinterpret_cast<const v2f*>(&GtabRe[zr * 16 + k0]);
    const v2f gim = *reinterpret_cast<const v2f*>(&GtabIm[zr * 16 + k0]);
    const v2f ngim = -gim;
    const float* pr = pr0 + k0 * 32;
    const float* pi = pi0 + k0 * 32;
    v2f br0, bi0, br1, bi1;
    br0.x = pr[m];      br0.y = pr[32 + m];
    bi0.x = pi[m];      bi0.y = pi[32 + m];
    br1.x = pr[16 + m]; br1.y = pr[48 + m];
    bi1.x = pi[16 + m]; bi1.y = pi[48 + m];
    a0 = wmma4(gre, br0, a0);
    a0 = wmma4(ngim, bi0, a0);  // Re(G*Y) = Gre*Yre - Gim*Yim
    a1 = wmma4(gre, br1, a1);
    a1 = wmma4(ngim, bi1, a1);
  }
  float* o = out + (size_t)site * 2048;
#pragma unroll
  for (int r = 0; r < 8; ++r) {
    const int zz = mtile * 16 + r + half * 8;
    o[zz * 32 + m] = a0[r] * IFFT_SCALE;
    o[zz * 32 + 16 + m] = a1[r] * IFFT_SCALE;
  }
}

// ---------------------------------------------------------------------------
extern "C" void kernel_launch(void* const* d_in, const int* in_sizes, int n_in,
                              void* d_out, int out_size, void* d_ws,
                              size_t ws_size, hipStream_t stream) {
  (void)in_sizes; (void)n_in; (void)out_size; (void)ws_size;
  const float* x = (const float*)d_in[0];       // [4][64][64][64][32]
  const float* wr = (const float*)d_in[1];      // [32][32][16][16][16]
  const float* wi = (const float*)d_in[2];
  float* out = (float*)d_out;                   // [4][64][64][64][32]
  float* ws = (float*)d_ws;

  // Workspace layout (floats). Big buffers reused across the symmetric halves.
  float* S1re = ws;                    // 8,388,608  [b][x][y][kz][i]
  float* S1im = S1re + 8388608;
  float* S2re = ws + 16777216;         // 2,097,152  [b][x][ky][kz][i]
  float* S2im = S2re + 2097152;
  float* S3re = ws + 20971520;         //   524,288  [b][kx][ky][kz][i]
  float* S3im = S3re + 524288;
  float* WTre = ws + 22020096;         // 4,194,304  [mode][i][o]
  float* WTim = WTre + 4194304;
  float* S4re = ws + 30408704;         //   524,288  [b][kx][ky][kz][o]
  float* S4im = S4re + 524288;
  float* S5re = S2re;                  // reuse: [b][x][ky][kz][o]
  float* S5im = S2im;
  float* S6re = S1re;                  // reuse: [b][x][y][kz][o]
  float* S6im = S1im;

  k_wt_transpose<<<16384, 256, 0, stream>>>(wr, wi, WTre, WTim);
  // forward pruned DFTs: z (TDM-fed), y, x
  k_fwd_z_real<<<4096, 128, 0, stream>>>(x, S1re, S1im);
  k_fwd_dft_cplx<<<1024, 256, 0, stream>>>(S1re, S1im, S2re, S2im, 512, 32);
  k_fwd_dft_cplx<<<256, 256, 0, stream>>>(S2re, S2im, S3re, S3im, 8192, 512);
  // per-mode channel mix
  k_mode_mix<<<512, 256, 0, stream>>>(S3re, S3im, WTre, WTim, S4re, S4im);
  // inverse expansions: x, y, z(+real part, scale)
  k_inv_dft_cplx<<<1024, 256, 0, stream>>>(S4re, S4im, S5re, S5im, 8192, 512);
  k_inv_dft_cplx<<<4096, 256, 0, stream>>>(S5re, S5im, S6re, S6im, 512, 32);
  k_inv_z_real<<<8192, 256, 0, stream>>>(S6re, S6im, out);
}